// MyModel_87522843560364
// MI455X (gfx1250) — compile-verified
//
#include <hip/hip_runtime.h>
#include <math.h>

// ---------------------------------------------------------------------------
// Bidirectional LSTM, B=128 T=2048 F=128 H=8.
// Phase 1: xz = x @ [W_fw | W_bw]  via V_WMMA_F32_16X16X4_F32 (fp32 WMMA),
//          stored time-major, gate-permuted: xz[dir][t][b][j*4+gate].
// Phase 2: sequential scan, 8 lanes per (batch,dir) chain, shfl-based h@U.
// ---------------------------------------------------------------------------

typedef __attribute__((ext_vector_type(2))) float v2f;
typedef __attribute__((ext_vector_type(8))) float v8f;

#define BB 128
#define TT 2048
#define FF 128

// ------------------------- Phase 1: WMMA projection -------------------------
// Grid: 2048 blocks x 256 threads (8 waves). Wave w handles M-tile
// mtile = blockIdx*8+w (16 rows of the 262144 x 128 "A" = x), N = 64 cols
// (W_fw 32 cols ++ W_bw 32 cols), K = 128 -> 32 wmma_f32_16x16x4_f32 per tile.
__global__ __launch_bounds__(256) void proj_gemm_wmma(
    const float* __restrict__ x,    // [B*T, 128]
    const float* __restrict__ Wfw,  // [128, 32]
    const float* __restrict__ Wbw,  // [128, 32]
    float* __restrict__ xz)         // [2, T, B, 32] permuted (j*4+gate)
{
    const int lane = threadIdx.x & 31;
    const int wave = threadIdx.x >> 5;
    const int mtile = blockIdx.x * 8 + wave;   // 0..16383
    const int m0 = mtile * 16;
    const int half = lane >> 4;                // 0: lanes 0-15, 1: lanes 16-31
    const int l15  = lane & 15;

    // A: lane holds row (m0+l15), K-pair base = half*2 (ISA 16x4 f32 layout)
    const float* xrow = x + (size_t)(m0 + l15) * FF + half * 2;
    const float* Wp[2] = {Wfw, Wbw};

    v8f acc[4] = {};   // nt = 0,1 -> fw cols [0:16),[16:32); nt = 2,3 -> bw

    for (int kk = 0; kk < 32; ++kk) {
        v2f a;
        a.x = xrow[kk * 4 + 0];   // K = 4kk + 2*half
        a.y = xrow[kk * 4 + 1];   // K = 4kk + 2*half + 1
        const int k0 = kk * 4 + half * 2;
#pragma unroll
        for (int nt = 0; nt < 4; ++nt) {
            const float* Wd = Wp[nt >> 1];
            const int col = (nt & 1) * 16 + l15;
            v2f bm;
            bm.x = Wd[(size_t)k0 * 32 + col];        // B row K=k0,  col N
            bm.y = Wd[(size_t)(k0 + 1) * 32 + col];  // B row K=k0+1
            acc[nt] = __builtin_amdgcn_wmma_f32_16x16x4_f32(
                false, a, false, bm, (short)0, acc[nt], false, false);
        }
    }

    // D layout: VGPR r -> lanes 0-15 = (M=r, N=l15), lanes 16-31 = (M=r+8).
    // Store gate-permuted: n_dir in [0,32) maps to p = (n&7)*4 + (n>>3)
    // so the scan kernel reads {i,f,g,o}[j] as one float4 at offset j*4.
#pragma unroll
    for (int nt = 0; nt < 4; ++nt) {
        const int dir  = nt >> 1;
        const int ndir = (nt & 1) * 16 + l15;        // col within direction
        const int p    = (ndir & 7) * 4 + (ndir >> 3);
#pragma unroll
        for (int r = 0; r < 8; ++r) {
            const int m = m0 + half * 8 + r;         // row = b*T + t
            const int b = m >> 11;                   // m / 2048
            const int t = m & 2047;                  // m % 2048
            const size_t idx =
                (((size_t)dir * TT + t) * BB + b) * 32 + p;
            xz[idx] = acc[nt][r];
        }
    }
}

// ------------------------- Phase 2: recurrent scan --------------------------
__device__ __forceinline__ float fsig(float x) {
    return 1.0f / (1.0f + __expf(-x));               // safe both tails
}
__device__ __forceinline__ float ftanh(float x) {
    const float a = fabsf(x);
    const float e = __expf(-2.0f * a);               // in (0,1], no overflow
    return copysignf((1.0f - e) / (1.0f + e), x);
}

// Grid: 64 blocks x 32 threads (one wave each). 256 chains (b,dir), 8 lanes
// per chain: lane j owns hidden unit j. h@U done with width-8 shuffles.
__global__ __launch_bounds__(32) void lstm_scan(
    const float* __restrict__ xz,   // [2, T, B, 32] permuted
    const float* __restrict__ Ufw, const float* __restrict__ bfw,
    const float* __restrict__ Ubw, const float* __restrict__ bbw,
    float* __restrict__ out)        // [B, T, 16]  (fw: 0-7, bw: 8-15)
{
    const int tid  = blockIdx.x * 32 + threadIdx.x;  // 0..2047
    const int elem = tid >> 3;                       // 0..255
    const int j    = tid & 7;
    const int b    = elem & (BB - 1);
    const int dir  = elem >> 7;

    const float* U    = dir ? Ubw : Ufw;
    const float* bias = dir ? bbw : bfw;

    float Ui[8], Uf[8], Ug[8], Uo[8];
#pragma unroll
    for (int k = 0; k < 8; ++k) {
        Ui[k] = U[k * 32 + j];
        Uf[k] = U[k * 32 + 8 + j];
        Ug[k] = U[k * 32 + 16 + j];
        Uo[k] = U[k * 32 + 24 + j];
    }
    const float bi = bias[j],      bf_ = bias[8 + j];
    const float bg = bias[16 + j], bo  = bias[24 + j];

    const int  t0    = dir ? (TT - 1) : 0;
    const long step  = dir ? -(long)(BB * 32) : (long)(BB * 32);
    const long ostep = dir ? -16L : 16L;

    const float* p  = xz + ((size_t)dir * TT + t0) * (BB * 32) + b * 32 + j * 4;
    float*       po = out + ((size_t)b * TT + t0) * 16 + dir * 8 + j;

    float h = 0.0f, c = 0.0f;
    float4 cur = *(const float4*)p;                  // t = t0

    for (int t = 0; t < TT; ++t) {
        // double-buffer: issue next timestep's load before the serial chain
        const float4 nxt = *(const float4*)(p + ((t + 1 < TT) ? step : 0));

        float zi = cur.x + bi, zf = cur.y + bf_;
        float zg = cur.z + bg, zo = cur.w + bo;
#pragma unroll
        for (int k = 0; k < 8; ++k) {
            const float hk = __shfl(h, k, 8);        // h[k] of this chain
            zi += hk * Ui[k];
            zf += hk * Uf[k];
            zg += hk * Ug[k];
            zo += hk * Uo[k];
        }
        const float ig = fsig(zi), fg = fsig(zf);
        const float gg = ftanh(zg), og = fsig(zo);
        c = fg * c + ig * gg;
        h = og * ftanh(c);

        *po = h;
        p  += step;
        po += ostep;
        cur = nxt;
    }
}

// ----------------------------------------------------------------------------
extern "C" void kernel_launch(void* const* d_in, const int* in_sizes, int n_in,
                              void* d_out, int out_size, void* d_ws, size_t ws_size,
                              hipStream_t stream) {
    (void)in_sizes; (void)n_in; (void)out_size; (void)ws_size;
    const float* x   = (const float*)d_in[0];
    const float* Wfw = (const float*)d_in[1];
    const float* Ufw = (const float*)d_in[2];
    const float* bfw = (const float*)d_in[3];
    const float* Wbw = (const float*)d_in[4];
    const float* Ubw = (const float*)d_in[5];
    const float* bbw = (const float*)d_in[6];
    float* out = (float*)d_out;
    float* xz  = (float*)d_ws;   // needs 2*2048*128*32*4 = 64 MiB

    // 262144/16 = 16384 M-tiles, 8 waves/block -> 2048 blocks
    proj_gemm_wmma<<<2048, 256, 0, stream>>>(x, Wfw, Wbw, xz);
    lstm_scan<<<64, 32, 0, stream>>>(xz, Ufw, bfw, Ubw, bbw, out);
}